// Wavelet_Convolution_53661321397055
// MI455X (gfx1250) — compile-verified
//
#include <hip/hip_runtime.h>

typedef float v2f __attribute__((ext_vector_type(2)));
typedef float v4f __attribute__((ext_vector_type(4)));
typedef float v8f __attribute__((ext_vector_type(8)));

#define F 128            // F_IN == F_OUT == 128
#define TILE_ROWS 128    // rows of x per workgroup
#define A_STRIDE 132     // pad 128 -> 132 floats: conflict-free A b64 reads
#define KP 64            // number of k-pairs (F/2)
#define WP_STRIDE 144    // k-pair row stride in float2 units; 144 mod 32 == 16
                         // -> lane halves hit disjoint LDS bank-pairs

// ---------------------------------------------------------------------------
// Dense projection: Xp = x @ W  using V_WMMA_F32_16X16X4_F32 (fp32 tensor op).
// Block = 256 threads = 8 waves; each wave computes 16 rows x 128 cols.
// W is staged k-pair-interleaved in LDS so each B fragment is ONE ds_load_b64
// straight into an even-aligned VGPR pair (kills the v_dual_mov repacking
// seen in round 1).
// ---------------------------------------------------------------------------
__global__ __launch_bounds__(256) void gemm_xw(const float* __restrict__ x,
                                               const float* __restrict__ W,
                                               float* __restrict__ Xp,
                                               int n) {
    __shared__ float Alds[TILE_ROWS * A_STRIDE];   // 67.6 KB
    __shared__ float Wlds[KP * WP_STRIDE * 2];     // 72 KB, float2-interleaved

    const int tid = threadIdx.x;
    const long long rowBase = (long long)blockIdx.x * TILE_ROWS;

    // Stage W: read coalesced float4 rows, scatter into k-pair interleaved
    // layout: Wlds[(kp*WP_STRIDE + n)*2 + (k&1)] = W[k][n].
    #pragma unroll
    for (int i = tid; i < (F * F) / 4; i += 256) {
        int k  = i >> 5;            // 32 float4 per row of W
        int c  = (i & 31) * 4;
        v4f w = *(const v4f*)(W + k * F + c);
        float* base = &Wlds[((k >> 1) * WP_STRIDE + c) * 2 + (k & 1)];
        base[0] = w.x; base[2] = w.y; base[4] = w.z; base[6] = w.w;
    }
    // Stage 128 rows of x into LDS with padded stride (row clamp for tail block).
    for (int i = tid; i < TILE_ROWS * (F / 4); i += 256) {
        int r  = i >> 5;            // 32 float4 per row
        int c4 = i & 31;
        long long gr = rowBase + r;
        if (gr >= n) gr = n - 1;    // clamp; stores are guarded below
        v4f val = *(const v4f*)(x + gr * F + c4 * 4);
        float* dst = &Alds[r * A_STRIDE + c4 * 4];
        dst[0] = val.x; dst[1] = val.y; dst[2] = val.z; dst[3] = val.w;
    }
    __syncthreads();

    const int wave = tid >> 5;
    const int lane = tid & 31;
    const int half = lane >> 4;     // 0: lanes 0-15, 1: lanes 16-31
    const int ln   = lane & 15;
    const int waveRow = wave * 16;

    v8f acc[8] = {};

    // A-fragment source row for this lane (M = ln).
    const float* arow = &Alds[(waveRow + ln) * A_STRIDE];

    for (int k = 0; k < F; k += 4) {
        // A 16x4 layout: lane half selects K {k,k+1} vs {k+2,k+3} -> one b64.
        v2f a = *(const v2f*)&arow[k + 2 * half];
        const int kp = (k >> 1) + half;       // k-pair row for this lane half
        #pragma unroll
        for (int t = 0; t < 8; ++t) {
            // B 4x16: (W[kk][n], W[kk+1][n]) adjacent -> one b64.
            v2f b = *(const v2f*)&Wlds[(kp * WP_STRIDE + t * 16 + ln) * 2];
            acc[t] = __builtin_amdgcn_wmma_f32_16x16x4_f32(
                /*neg_a=*/false, a, /*neg_b=*/false, b,
                /*c_mod=*/(short)0, acc[t],
                /*reuse_a=*/false, /*reuse_b=*/false);
        }
    }

    // C/D layout: VGPR r holds M=r (lanes 0-15) and M=r+8 (lanes 16-31).
    #pragma unroll
    for (int t = 0; t < 8; ++t) {
        #pragma unroll
        for (int r = 0; r < 8; ++r) {
            long long row = rowBase + waveRow + r + half * 8;
            if (row < n) Xp[row * F + t * 16 + ln] = acc[t][r];
        }
    }
}

// ---------------------------------------------------------------------------
// COO SpMM: dst[rows[e], :] += vals[e] * (rowScale?[cols[e]]) * src[cols[e], :]
// One wave per nonzero; lane handles a float4 slice (32 lanes * 4 = 128).
// src/dst are L2-resident (51 MB each << 192 MB L2); the one-shot COO stream
// is read non-temporally so it doesn't evict the hot feature matrices.
// ---------------------------------------------------------------------------
__global__ __launch_bounds__(256) void spmm_coo(const long long* __restrict__ rows,
                                                const long long* __restrict__ cols,
                                                const float* __restrict__ vals,
                                                const float* __restrict__ rowScale,
                                                const float* __restrict__ src,
                                                float* __restrict__ dst,
                                                long long nnz) {
    const int lane = threadIdx.x & 31;
    const long long wavesTotal = (long long)gridDim.x * (blockDim.x >> 5);
    long long w = (long long)blockIdx.x * (blockDim.x >> 5) + (threadIdx.x >> 5);

    for (long long e = w; e < nnz; e += wavesTotal) {
        long long r = __builtin_nontemporal_load(rows + e);
        long long c = __builtin_nontemporal_load(cols + e);
        float v = __builtin_nontemporal_load(vals + e);
        if (rowScale) v *= rowScale[c];   // kernel * X_t scaling (phi1 pass)
        v4f xv = *(const v4f*)(src + c * F + lane * 4);
        float* d = dst + r * F + lane * 4;
        atomicAdd(d + 0, v * xv.x);
        atomicAdd(d + 1, v * xv.y);
        atomicAdd(d + 2, v * xv.z);
        atomicAdd(d + 3, v * xv.w);
    }
}

// ---------------------------------------------------------------------------
// In-place relu over the accumulated output.
// ---------------------------------------------------------------------------
__global__ __launch_bounds__(256) void relu_inplace(float* __restrict__ out,
                                                    long long n4) {
    long long i = (long long)blockIdx.x * blockDim.x + threadIdx.x;
    if (i < n4) {
        v4f v = ((v4f*)out)[i];
        v.x = fmaxf(v.x, 0.f);
        v.y = fmaxf(v.y, 0.f);
        v.z = fmaxf(v.z, 0.f);
        v.w = fmaxf(v.w, 0.f);
        ((v4f*)out)[i] = v;
    }
}

extern "C" void kernel_launch(void* const* d_in, const int* in_sizes, int n_in,
                              void* d_out, int out_size, void* d_ws, size_t ws_size,
                              hipStream_t stream) {
    const float*      x    = (const float*)d_in[0];
    const float*      W    = (const float*)d_in[1];
    const float*      kern = (const float*)d_in[2];           // ones(N,1)
    const long long*  p0r  = (const long long*)d_in[3];
    const long long*  p0c  = (const long long*)d_in[4];
    const float*      p0v  = (const float*)d_in[5];
    const long long*  p1r  = (const long long*)d_in[6];
    const long long*  p1c  = (const long long*)d_in[7];
    const float*      p1v  = (const float*)d_in[8];

    const int       n    = in_sizes[0] / F;                   // 100000
    const long long nnz0 = in_sizes[5];
    const long long nnz1 = in_sizes[8];

    float* out = (float*)d_out;       // doubles as Xp buffer, then final out
    float* Xt  = (float*)d_ws;        // N*F floats of workspace

    const size_t featBytes = (size_t)n * F * sizeof(float);

    // 1) Xt := 0 (workspace is poisoned by the harness)
    hipMemsetAsync(Xt, 0, featBytes, stream);

    // 2) Xp = x @ W   (written into d_out as scratch)
    gemm_xw<<<(n + TILE_ROWS - 1) / TILE_ROWS, 256, 0, stream>>>(x, W, out, n);

    // 3) Xt += phi0 @ Xp
    {
        long long blocks = (nnz0 + 7) / 8;                    // 8 waves / block
        spmm_coo<<<(unsigned)blocks, 256, 0, stream>>>(p0r, p0c, p0v, nullptr,
                                                       out, Xt, nnz0);
    }

    // 4) out := 0, then out += phi1 @ (kernel ⊙ Xt)
    hipMemsetAsync(out, 0, featBytes, stream);
    {
        long long blocks = (nnz1 + 7) / 8;
        spmm_coo<<<(unsigned)blocks, 256, 0, stream>>>(p1r, p1c, p1v, kern,
                                                       Xt, out, nnz1);
    }

    // 5) relu in place
    {
        long long n4 = (long long)n * F / 4;
        long long blocks = (n4 + 255) / 256;
        relu_inplace<<<(unsigned)blocks, 256, 0, stream>>>(out, n4);
    }
}